// DynamicPFNLayer_17454747091076
// MI455X (gfx1250) — compile-verified
//
#include <hip/hip_runtime.h>
#include <hip/hip_bf16.h>

typedef __attribute__((ext_vector_type(2))) float v2f;
typedef __attribute__((ext_vector_type(8))) float v8f;

#define BN_EPS 1e-3f

// ---------------------------------------------------------------- init ws
__global__ void k_init(float* __restrict__ gsum, float* __restrict__ gsq,
                       float* __restrict__ feat, const int* __restrict__ nseg) {
  long long total  = (long long)(*nseg) * 64;
  long long stride = (long long)gridDim.x * blockDim.x;
  long long gid    = (long long)blockIdx.x * blockDim.x + threadIdx.x;
  for (long long i = gid; i < total; i += stride) feat[i] = 0.0f;
  if (gid < 64) { gsum[gid] = 0.0f; gsq[gid] = 0.0f; }
}

// --------------------------------------- WMMA f32 GEMM + channel statistics
// x = inputs @ W^T  (x[n,u] = sum_c in[n,c]*W[u,c]), raw x staged in out[:, :64]
__global__ __launch_bounds__(256)
void k_gemm_stats(const float* __restrict__ in, const float* __restrict__ Wm,
                  float* __restrict__ out, float* __restrict__ gsum,
                  float* __restrict__ gsq, int N) {
  __shared__ float lA[128 * 64];   // 128-row input tile
  __shared__ float lW[64 * 64];    // full weight matrix [u][c]
  __shared__ float lsum[64];
  __shared__ float lsq[64];

  const int tid = threadIdx.x;
  const long long rowBase = (long long)blockIdx.x * 128;

  for (int i = tid; i < 1024; i += 256)                 // 64*64 floats
    ((float4*)lW)[i] = ((const float4*)Wm)[i];
  for (int i = tid; i < 2048; i += 256) {               // 128*64 floats
    long long r = rowBase + (i >> 4);
    if (r > (long long)N - 1) r = (long long)N - 1;     // clamp partial tile
    ((float4*)lA)[i] = ((const float4*)in)[r * 16 + (i & 15)];
  }
  if (tid < 64) { lsum[tid] = 0.0f; lsq[tid] = 0.0f; }
  __syncthreads();

  const int lane = tid & 31;
  const int wave = tid >> 5;
  const int l15  = lane & 15;
  const int half = lane >> 4;     // 0: lanes 0-15, 1: lanes 16-31
  const int rloc = wave * 16;     // this wave's 16-row sub-tile

  v8f acc[4] = {};                // 4 accumulators -> units [0..63]
  const float* aRow = &lA[(rloc + l15) * 64];
  const float* bp0  = &lW[(l15 +  0) * 64];
  const float* bp1  = &lW[(l15 + 16) * 64];
  const float* bp2  = &lW[(l15 + 32) * 64];
  const float* bp3  = &lW[(l15 + 48) * 64];

#pragma unroll
  for (int k0 = 0; k0 < 64; k0 += 4) {
    const int kk = k0 + 2 * half;                 // K pair handled by this lane half
    v2f a  = *(const v2f*)&aRow[kk];              // A 16x4 frag (ISA layout)
    v2f b0 = *(const v2f*)&bp0[kk];               // B 4x16 frags: B[k][u] = W[u][k]
    v2f b1 = *(const v2f*)&bp1[kk];
    v2f b2 = *(const v2f*)&bp2[kk];
    v2f b3 = *(const v2f*)&bp3[kk];
    acc[0] = __builtin_amdgcn_wmma_f32_16x16x4_f32(false, a, false, b0, (short)0, acc[0], false, false);
    acc[1] = __builtin_amdgcn_wmma_f32_16x16x4_f32(false, a, false, b1, (short)0, acc[1], false, false);
    acc[2] = __builtin_amdgcn_wmma_f32_16x16x4_f32(false, a, false, b2, (short)0, acc[2], false, false);
    acc[3] = __builtin_amdgcn_wmma_f32_16x16x4_f32(false, a, false, b3, (short)0, acc[3], false, false);
  }

  // store raw x into out[:, :64] and accumulate per-channel sum / sum-of-squares
#pragma unroll
  for (int j = 0; j < 4; ++j) {
    float ps = 0.0f, pq = 0.0f;
#pragma unroll
    for (int i = 0; i < 8; ++i) {                 // C/D layout: VGPR i -> row i (+8 for half 1)
      long long row = rowBase + rloc + i + 8 * half;
      if (row < (long long)N) {
        float v = acc[j][i];
        out[row * 128 + 16 * j + l15] = v;
        ps += v; pq += v * v;
      }
    }
    atomicAdd(&lsum[16 * j + l15], ps);
    atomicAdd(&lsq [16 * j + l15], pq);
  }
  __syncthreads();
  if (tid < 64) {
    atomicAdd(&gsum[tid], lsum[tid]);
    atomicAdd(&gsq [tid], lsq [tid]);
  }
}

// -------------------------------------------------- fold BN into scale/shift
__global__ void k_stats(const float* __restrict__ gsum, const float* __restrict__ gsq,
                        const float* __restrict__ gamma, const float* __restrict__ beta,
                        float* __restrict__ scale, float* __restrict__ shift, int N) {
  int u = threadIdx.x;                                  // blockDim = 64
  float invN = 1.0f / (float)N;
  float m    = gsum[u] * invN;
  float var  = gsq[u] * invN - m * m;                   // biased variance
  float sc   = gamma[u] * rsqrtf(var + BN_EPS);
  scale[u] = sc;
  shift[u] = beta[u] - m * sc;
}

// --------------------------- BN + ReLU in place, scatter-max into feat_max
__global__ __launch_bounds__(256)
void k_bnrelu_scmax(float* __restrict__ out, const long long* __restrict__ unq,
                    const float* __restrict__ scale, const float* __restrict__ shift,
                    unsigned* __restrict__ featb, long long total) {
  __shared__ float ls[64], lh[64];
  if (threadIdx.x < 64) { ls[threadIdx.x] = scale[threadIdx.x]; lh[threadIdx.x] = shift[threadIdx.x]; }
  __syncthreads();
  long long g = (long long)blockIdx.x * 256 + threadIdx.x;
  if (g >= total) return;
  long long n = g >> 4;
  int q = (int)(g & 15), c = q * 4;
  float4 x = ((float4*)out)[n * 32 + q];                // row = 32 float4; first 16 = x
  float4 y;
  y.x = fmaxf(0.0f, fmaf(x.x, ls[c + 0], lh[c + 0]));
  y.y = fmaxf(0.0f, fmaf(x.y, ls[c + 1], lh[c + 1]));
  y.z = fmaxf(0.0f, fmaf(x.z, ls[c + 2], lh[c + 2]));
  y.w = fmaxf(0.0f, fmaf(x.w, ls[c + 3], lh[c + 3]));
  ((float4*)out)[n * 32 + q] = y;
  long long seg = unq[n];
  unsigned* b = featb + seg * 64 + c;                   // y >= 0: uint bits order == float order
  atomicMax(b + 0, __float_as_uint(y.x));
  atomicMax(b + 1, __float_as_uint(y.y));
  atomicMax(b + 2, __float_as_uint(y.z));
  atomicMax(b + 3, __float_as_uint(y.w));
}

// -------------------------------------------------- gather pillar max back
__global__ __launch_bounds__(256)
void k_gather(float* __restrict__ out, const long long* __restrict__ unq,
              const float* __restrict__ feat, long long total) {
  long long g = (long long)blockIdx.x * 256 + threadIdx.x;
  if (g >= total) return;
  long long n = g >> 4;
  int q = (int)(g & 15);
  long long seg = unq[n];
  float4 v = ((const float4*)feat)[seg * 16 + q];
  ((float4*)out)[n * 32 + 16 + q] = v;
}

extern "C" void kernel_launch(void* const* d_in, const int* in_sizes, int n_in,
                              void* d_out, int out_size, void* d_ws, size_t ws_size,
                              hipStream_t stream) {
  const float*     in    = (const float*)d_in[0];
  const long long* unq   = (const long long*)d_in[1];   // jnp.int64
  const int*       nseg  = (const int*)d_in[2];         // scalar on device
  const float*     Wm    = (const float*)d_in[3];
  const float*     gamma = (const float*)d_in[4];
  const float*     beta  = (const float*)d_in[5];
  float*           out   = (float*)d_out;

  const int N = in_sizes[0] / 64;

  float* gsum  = (float*)d_ws;        // [64]
  float* gsq   = gsum + 64;           // [64]
  float* scale = gsq + 64;            // [64]
  float* shift = scale + 64;          // [64]
  float* feat  = shift + 64;          // [num_segments * 64]

  k_init<<<4096, 256, 0, stream>>>(gsum, gsq, feat, nseg);

  int gblocks = (N + 127) / 128;
  k_gemm_stats<<<gblocks, 256, 0, stream>>>(in, Wm, out, gsum, gsq, N);

  k_stats<<<1, 64, 0, stream>>>(gsum, gsq, gamma, beta, scale, shift, N);

  long long total = (long long)N * 16;
  int eblocks = (int)((total + 255) / 256);
  k_bnrelu_scmax<<<eblocks, 256, 0, stream>>>(out, unq, scale, shift, (unsigned*)feat, total);
  k_gather<<<eblocks, 256, 0, stream>>>(out, unq, feat, total);
}